// TemporalUserPair_29334626632394
// MI455X (gfx1250) — compile-verified
//
#include <hip/hip_runtime.h>
#include <math.h>

typedef __attribute__((ext_vector_type(2))) float v2f;
typedef __attribute__((ext_vector_type(8))) float v8f;

#define BASE_DIM 512
#define NUM_HEADS 4
#define HEAD_DIM 128
#define TIMESLOTS 24
#define BB 64
#define SS 512
#define NN (BB * SS)          // 32768 rows
#define KDIM 96               // GEMM K = NUM_HEADS * TIMESLOTS
#define NKK 24                // KDIM / 4 wmma k-steps
#define NCT 34                // col tiles: 32 (at_emb 512) + 2 (logits padded to 32)
#define A_STRIDE 98           // LDS row stride (even, conflict-avoiding)

// ---- workspace layout (float offsets) ----
#define WS_KMAT 0
#define WS_VMAT (WS_KMAT + TIMESLOTS * BASE_DIM)            // 12288
#define WS_QU   (WS_VMAT + TIMESLOTS * BASE_DIM)            // 24576
#define WS_QT   (WS_QU + BB * BASE_DIM)                     // 57344
#define WS_SU   (WS_QT + TIMESLOTS * BASE_DIM)              // 69632
#define WS_ST   (WS_SU + BB * KDIM)                         // 75776
#define WS_VUP  (WS_ST + TIMESLOTS * KDIM)                  // 78080
// total = 78080 + 24*34*64 = 130304 floats (~521 KB)

// ---------------------------------------------------------------------------
// Kernel 1: projections.
//   kmat[t, he] = tse[t,:] . Wk[he,:] + bk        (he = h*128+e, flat 512)
//   vmat[t, he] = tse[t,:] . Wv[he,:] + bv
//   qu[b, he]   = upt[user_x[b],:] . Wq[he, 0:512]
//   qt[t, he]   = tse[t,:] . Wq[he, 512:1024] + bq
// 136 rows x 512 cols, one thread per element.
// ---------------------------------------------------------------------------
__global__ __launch_bounds__(256)
void tup_proj_kernel(const float* __restrict__ tse, const int* __restrict__ user_x,
                     const float* __restrict__ upt,
                     const float* __restrict__ Wq, const float* __restrict__ bq,
                     const float* __restrict__ Wk, const float* __restrict__ bk,
                     const float* __restrict__ Wv, const float* __restrict__ bv,
                     float* __restrict__ ws) {
  int id = blockIdx.x * blockDim.x + threadIdx.x;
  if (id >= 136 * BASE_DIM) return;
  int he = id % BASE_DIM;
  int r  = id / BASE_DIM;
  float acc = 0.0f;
  if (r < 24) {                                   // kmat
    const float* x = tse + r * BASE_DIM;
    const float* w = Wk + (size_t)he * BASE_DIM;
#pragma unroll 8
    for (int d = 0; d < BASE_DIM; ++d) acc += x[d] * w[d];
    ws[WS_KMAT + r * BASE_DIM + he] = acc + bk[he];
  } else if (r < 48) {                            // vmat
    int t = r - 24;
    const float* x = tse + t * BASE_DIM;
    const float* w = Wv + (size_t)he * BASE_DIM;
#pragma unroll 8
    for (int d = 0; d < BASE_DIM; ++d) acc += x[d] * w[d];
    ws[WS_VMAT + t * BASE_DIM + he] = acc + bv[he];
  } else if (r < 112) {                           // qu
    int b = r - 48;
    const float* x = upt + (size_t)user_x[b] * BASE_DIM;
    const float* w = Wq + (size_t)he * (2 * BASE_DIM);
#pragma unroll 8
    for (int d = 0; d < BASE_DIM; ++d) acc += x[d] * w[d];
    ws[WS_QU + b * BASE_DIM + he] = acc;
  } else {                                        // qt (includes bq)
    int t = r - 112;
    const float* x = tse + t * BASE_DIM;
    const float* w = Wq + (size_t)he * (2 * BASE_DIM) + BASE_DIM;
#pragma unroll 8
    for (int d = 0; d < BASE_DIM; ++d) acc += x[d] * w[d];
    ws[WS_QT + t * BASE_DIM + he] = acc + bq[he];
  }
}

// ---------------------------------------------------------------------------
// Kernel 2: score tables + packed B matrix.
//   VUp: B fragments for v_wmma_f32_16x16x4, packed so lane l of step (kk,ct)
//        loads one contiguous float2:
//        VUp[((kk*34+ct)*32 + l)*2 + p] = VU[kk*4 + 2*(l>>4) + p][colbase + (l&15)]
//        VU[r, j] = vmat[r%24, (r/24)*128 + e] . W_unify[j, (r/24)*128 + e]
//        (ct>=32 -> W_time columns, zero-padded past 24)
//   su[b, r]  = qu[b, h*128:] . kmat[ks, h*128:]     (r = h*24+ks)
//   st[t, r]  = qt[t, h*128:] . kmat[ks, h*128:]
// ---------------------------------------------------------------------------
__global__ __launch_bounds__(256)
void tup_tables_kernel(const float* __restrict__ W_unify,
                       const float* __restrict__ W_time,
                       float* __restrict__ ws) {
  const int NVUP = NKK * NCT * 64;                // 52224
  int id = blockIdx.x * blockDim.x + threadIdx.x;
  if (id < NVUP) {
    int p  = id & 1;
    int l  = (id >> 1) & 31;
    int ck = id >> 6;                             // kk*34 + ct
    int ct = ck % NCT;
    int kk = ck / NCT;
    int r  = kk * 4 + 2 * (l >> 4) + p;           // 0..95
    int h  = r / TIMESLOTS;
    int ks = r % TIMESLOTS;
    const float* v = ws + WS_VMAT + ks * BASE_DIM + h * HEAD_DIM;
    float acc = 0.0f;
    if (ct < 32) {
      int j = ct * 16 + (l & 15);
      const float* w = W_unify + (size_t)j * BASE_DIM + h * HEAD_DIM;
#pragma unroll 8
      for (int e = 0; e < HEAD_DIM; ++e) acc += v[e] * w[e];
    } else {
      int j = (ct - 32) * 16 + (l & 15);
      if (j < TIMESLOTS) {
        const float* w = W_time + (size_t)j * BASE_DIM + h * HEAD_DIM;
#pragma unroll 8
        for (int e = 0; e < HEAD_DIM; ++e) acc += v[e] * w[e];
      }
    }
    ws[WS_VUP + id] = acc;
  } else if (id < NVUP + BB * KDIM) {             // su
    int s = id - NVUP;
    int b = s / KDIM, r = s % KDIM;
    int h = r / TIMESLOTS, ks = r % TIMESLOTS;
    const float* q = ws + WS_QU + b * BASE_DIM + h * HEAD_DIM;
    const float* k = ws + WS_KMAT + ks * BASE_DIM + h * HEAD_DIM;
    float acc = 0.0f;
#pragma unroll 8
    for (int e = 0; e < HEAD_DIM; ++e) acc += q[e] * k[e];
    ws[WS_SU + s] = acc;
  } else if (id < NVUP + BB * KDIM + TIMESLOTS * KDIM) {  // st
    int s = id - NVUP - BB * KDIM;
    int t = s / KDIM, r = s % KDIM;
    int h = r / TIMESLOTS, ks = r % TIMESLOTS;
    const float* q = ws + WS_QT + t * BASE_DIM + h * HEAD_DIM;
    const float* k = ws + WS_KMAT + ks * BASE_DIM + h * HEAD_DIM;
    float acc = 0.0f;
#pragma unroll 8
    for (int e = 0; e < HEAD_DIM; ++e) acc += q[e] * k[e];
    ws[WS_ST + s] = acc;
  }
}

// ---------------------------------------------------------------------------
// Kernel 3: per-row masked softmax + fused A[32,96] x VUp[96, 512|24] via WMMA.
// 1024 blocks x 128 threads (4 wave32). Block owns 32 rows of N.
// ---------------------------------------------------------------------------
__global__ __launch_bounds__(128)
void tup_attn_gemm_kernel(const int* __restrict__ hour_x,
                          const int* __restrict__ hour_mask,
                          const float* __restrict__ ws,
                          const float* __restrict__ b_unify,
                          const float* __restrict__ b_time,
                          float* __restrict__ out) {
  __shared__ float Alds[32 * A_STRIDE];

  const int t    = threadIdx.x;
  const int rr   = t >> 2;                 // 0..31 row within block
  const int h    = t & 3;                  // head
  const int n    = blockIdx.x * 32 + rr;   // global row
  const int b    = n >> 9;                 // n / SS
  const int hr   = hour_x[n];
  const float scale = 0.08838834764831845f;  // 1/sqrt(128)

  // ---- phase 1: masked softmax over 24 slots for (row, head) ----
  {
    const float* su = ws + WS_SU + b * KDIM + h * TIMESLOTS;
    const float* st = ws + WS_ST + hr * KDIM + h * TIMESLOTS;
    const int*   mk = hour_mask + (size_t)n * TIMESLOTS;
    float s[TIMESLOTS];
    float mx = -__builtin_inff();
#pragma unroll
    for (int k = 0; k < TIMESLOTS; ++k) {
      float v = (mk[k] == 1) ? -__builtin_inff() : (su[k] + st[k]) * scale;
      s[k] = v;
      mx = fmaxf(mx, v);
    }
    float sum = 0.0f;
#pragma unroll
    for (int k = 0; k < TIMESLOTS; ++k) {
      float e = __expf(s[k] - mx);
      s[k] = e;
      sum += e;
    }
    float inv = 1.0f / sum;
#pragma unroll
    for (int k = 0; k < TIMESLOTS; ++k)
      Alds[rr * A_STRIDE + h * TIMESLOTS + k] = s[k] * inv;
  }
  __syncthreads();

  // ---- phase 2: WMMA GEMM. 68 tiles (2 row-tiles x 34 col-tiles), 17/wave ----
  const int wave  = t >> 5;
  const int lane  = t & 31;
  const int m16   = lane & 15;
  const int khalf = lane >> 4;
  const float* vup = ws + WS_VUP;

  for (int i = 0; i < 17; ++i) {
    int tile = 4 * i + wave;               // uniform per wave
    int mt = tile & 1;
    int ct = tile >> 1;

    const float* arow = &Alds[(mt * 16 + m16) * A_STRIDE + 2 * khalf];
    const float* brow = vup + ct * 64 + lane * 2;

    v8f acc = {0.f, 0.f, 0.f, 0.f, 0.f, 0.f, 0.f, 0.f};
#pragma unroll 4
    for (int kk = 0; kk < NKK; ++kk) {
      v2f a = *(const v2f*)(arow + kk * 4);
      v2f bb = *(const v2f*)(brow + (size_t)kk * (NCT * 64));
      acc = __builtin_amdgcn_wmma_f32_16x16x4_f32(
          false, a, false, bb, (short)0, acc, false, false);
    }

    // epilogue: D layout — acc[q] is (row = mt*16 + 8*khalf + q, col = m16)
    int rowbase = blockIdx.x * 32 + mt * 16 + 8 * khalf;
    if (ct < 32) {
      int j = ct * 16 + m16;
      float bias = b_unify[j];
#pragma unroll
      for (int q = 0; q < 8; ++q)
        out[(size_t)(rowbase + q) * BASE_DIM + j] = acc[q] + bias;
    } else {
      int j = (ct - 32) * 16 + m16;
      if (j < TIMESLOTS) {
        float bias = b_time[j];
#pragma unroll
        for (int q = 0; q < 8; ++q)
          out[(size_t)NN * BASE_DIM + (size_t)(rowbase + q) * TIMESLOTS + j] =
              acc[q] + bias;
      }
    }
  }
}

// ---------------------------------------------------------------------------
extern "C" void kernel_launch(void* const* d_in, const int* in_sizes, int n_in,
                              void* d_out, int out_size, void* d_ws, size_t ws_size,
                              hipStream_t stream) {
  const float* tse      = (const float*)d_in[0];   // timeslot_embedded [24,512]
  // d_in[1] smoothed_timeslot_embedded: unused by reference
  // d_in[2] user_embedded: unused by reference
  const int*   user_x   = (const int*)d_in[3];     // [64]
  const int*   hour_x   = (const int*)d_in[4];     // [64,512]
  const int*   hour_mask= (const int*)d_in[5];     // [64,512,24]
  const float* upt      = (const float*)d_in[6];   // user_pref_table [100000,512]
  const float* Wq       = (const float*)d_in[7];   // [4,128,1024]
  const float* bq       = (const float*)d_in[8];   // [4,128]
  const float* Wk       = (const float*)d_in[9];   // [4,128,512]
  const float* bk       = (const float*)d_in[10];
  const float* Wv       = (const float*)d_in[11];
  const float* bv       = (const float*)d_in[12];
  const float* W_unify  = (const float*)d_in[13];  // [512,512]
  const float* b_unify  = (const float*)d_in[14];
  const float* W_time   = (const float*)d_in[15];  // [24,512]
  const float* b_time   = (const float*)d_in[16];
  float* ws  = (float*)d_ws;
  float* out = (float*)d_out;

  (void)in_sizes; (void)n_in; (void)out_size; (void)ws_size;

  // 1) projection tables: 136*512 elements
  tup_proj_kernel<<<(136 * BASE_DIM + 255) / 256, 256, 0, stream>>>(
      tse, user_x, upt, Wq, bq, Wk, bk, Wv, bv, ws);

  // 2) score tables + packed WMMA B matrix: 52224 + 6144 + 2304 = 60672 jobs
  tup_tables_kernel<<<(60672 + 255) / 256, 256, 0, stream>>>(W_unify, W_time, ws);

  // 3) masked softmax + fused output GEMM via V_WMMA_F32_16X16X4_F32
  tup_attn_gemm_kernel<<<NN / 32, 128, 0, stream>>>(
      hour_x, hour_mask, ws, b_unify, b_time, out);
}